// GraphLayer_7438883356900
// MI455X (gfx1250) — compile-verified
//
#include <hip/hip_runtime.h>
#include <hip/hip_bf16.h>

#define IN_F  128
#define OUT_F 64
#define BN_EPS 1e-5f

typedef float v2f __attribute__((ext_vector_type(2)));
typedef float v8f __attribute__((ext_vector_type(8)));

// ---- ordered-uint encoding for float atomicMax (monotone bijection) ----
__device__ __forceinline__ unsigned f2ord(float f) {
  unsigned b = __float_as_uint(f);
  return (b & 0x80000000u) ? ~b : (b | 0x80000000u);
}
__device__ __forceinline__ float ord2f(unsigned u) {
  unsigned b = (u & 0x80000000u) ? (u ^ 0x80000000u) : ~u;
  return __uint_as_float(b);
}

// ---- 0: zero accumulators (d_out is the agg accumulator) ----
__global__ __launch_bounds__(256) void k_init(float* __restrict__ out,
                                              unsigned* __restrict__ mxkey,
                                              float* __restrict__ ssum,
                                              float* __restrict__ colsum,
                                              float* __restrict__ colsq, int n) {
  long i = (long)blockIdx.x * blockDim.x + threadIdx.x;
  long total = (long)n * OUT_F;
  if (i < total) out[i] = 0.0f;
  if (i < n) { mxkey[i] = 0u; ssum[i] = 0.0f; }   // key 0 == -NaN, below all reals
  if (i < OUT_F) { colsum[i] = 0.0f; colsq[i] = 0.0f; }
}

// ---- 1: h = x @ W^T via v_wmma_f32_16x16x4_f32 ----
// One wave computes a 16x64 block: 4 col-tiles share the A fragment.
__global__ __launch_bounds__(256)
void k_gemm_wmma(const float* __restrict__ x, const float* __restrict__ W,
                 float* __restrict__ h, int nrows) {
  __shared__ float Wt[IN_F * OUT_F];                 // Wt[k][o] = W[o][k] (32 KB)
  for (int i = threadIdx.x; i < IN_F * OUT_F; i += blockDim.x) {
    int o = i >> 7;          // i / 128
    int k = i & (IN_F - 1);  // i % 128
    Wt[k * OUT_F + o] = W[i];
  }
  __syncthreads();

  const int lane = threadIdx.x & 31;
  const int wid  = blockIdx.x * (blockDim.x >> 5) + (threadIdx.x >> 5);
  const int m0   = wid * 16;
  if (m0 >= nrows) return;                          // wave-uniform exit

  const int half = lane >> 4;                       // 0: lanes 0-15, 1: 16-31
  const int l16  = lane & 15;
  long arow = m0 + l16;                             // A: M = lane&15 in both halves
  if (arow >= nrows) arow = nrows - 1;              // safety clamp (N%16==0 normally)

  v8f acc[4];
  #pragma unroll
  for (int t = 0; t < 4; ++t) acc[t] = v8f{0,0,0,0,0,0,0,0};

  for (int k = 0; k < IN_F; k += 4) {
    // A fragment 16x4: lanes 0-15 hold K=k+0/k+1, lanes 16-31 hold K=k+2/k+3
    const float* ap = x + arow * IN_F + k + half * 2;
    v2f a; a.x = ap[0]; a.y = ap[1];
    const int kk = k + half * 2;
    #pragma unroll
    for (int t = 0; t < 4; ++t) {
      const int col = t * 16 + l16;
      // B fragment 4x16: VGPR0 = row K=kk, VGPR1 = row K=kk+1, N = lane&15
      v2f b;
      b.x = Wt[(kk + 0) * OUT_F + col];
      b.y = Wt[(kk + 1) * OUT_F + col];
      acc[t] = __builtin_amdgcn_wmma_f32_16x16x4_f32(
          false, a, false, b, (short)0, acc[t], false, false);
    }
  }
  #pragma unroll
  for (int t = 0; t < 4; ++t) {
    #pragma unroll
    for (int j = 0; j < 8; ++j) {                   // C/D: VGPR j -> M=j (lo) / j+8 (hi)
      int row = m0 + j + half * 8;
      if (row < nrows) h[(long)row * OUT_F + t * 16 + l16] = acc[t][j];
    }
  }
}

// ---- 2: per-node attention scalars a_i, a_j (wave per node) ----
__global__ __launch_bounds__(256)
void k_scores(const float* __restrict__ h, const float* __restrict__ emb,
              const float* __restrict__ att_i, const float* __restrict__ att_j,
              const float* __restrict__ att_em_i, const float* __restrict__ att_em_j,
              float* __restrict__ a_i, float* __restrict__ a_j, int n) {
  const int lane = threadIdx.x & 31;
  const int node = blockIdx.x * (blockDim.x >> 5) + (threadIdx.x >> 5);
  if (node >= n) return;
  const long base = (long)node * OUT_F;
  float h0 = h[base + lane],      h1 = h[base + 32 + lane];
  float e0 = emb[base + lane],    e1 = emb[base + 32 + lane];
  float si = h0 * att_i[lane] + h1 * att_i[32 + lane]
           + e0 * att_em_i[lane] + e1 * att_em_i[32 + lane];
  float sj = h0 * att_j[lane] + h1 * att_j[32 + lane]
           + e0 * att_em_j[lane] + e1 * att_em_j[32 + lane];
  #pragma unroll
  for (int off = 16; off > 0; off >>= 1) {
    si += __shfl_xor(si, off, 32);
    sj += __shfl_xor(sj, off, 32);
  }
  if (lane == 0) { a_i[node] = si; a_j[node] = sj; }
}

// ---- 3: per-edge logits + segment max (ordered-uint atomicMax) ----
__global__ __launch_bounds__(256)
void k_edge_logit_max(const int* __restrict__ src, const int* __restrict__ dst,
                      const float* __restrict__ a_i, const float* __restrict__ a_j,
                      float* __restrict__ logits, unsigned* __restrict__ mxkey, long E) {
  long e = (long)blockIdx.x * blockDim.x + threadIdx.x;
  if (e >= E) return;
  int s = src[e], d = dst[e];
  float v = a_i[d] + a_j[s];
  v = (v >= 0.0f) ? v : 0.2f * v;                   // leaky_relu(., 0.2)
  logits[e] = v;
  atomicMax(&mxkey[d], f2ord(v));
}

// ---- 4: e = exp(logit - max); segment sum ----
__global__ __launch_bounds__(256)
void k_edge_expsum(const int* __restrict__ dst, float* logits /*in/out: exp*/,
                   const unsigned* __restrict__ mxkey, float* __restrict__ ssum, long E) {
  long e = (long)blockIdx.x * blockDim.x + threadIdx.x;
  if (e >= E) return;
  int d = dst[e];
  float ex = __expf(logits[e] - ord2f(mxkey[d]));
  logits[e] = ex;
  atomicAdd(&ssum[d], ex);
}

// ---- 5: agg[dst] += alpha * h[src]  (wave per edge; lane -> cols lane, lane+32) ----
__global__ __launch_bounds__(256)
void k_edge_agg(const int* __restrict__ src, const int* __restrict__ dst,
                const float* __restrict__ expv, const float* __restrict__ ssum,
                const float* __restrict__ h, float* __restrict__ out, long E) {
  const int lane = threadIdx.x & 31;
  long e = (long)blockIdx.x * (blockDim.x >> 5) + (threadIdx.x >> 5);
  if (e >= E) return;
  int s = src[e], d = dst[e];
  float alpha = expv[e] / (ssum[d] + 1e-16f);
  long hb = (long)s * OUT_F;
  long ob = (long)d * OUT_F;
  atomicAdd(&out[ob + lane],      alpha * h[hb + lane]);
  atomicAdd(&out[ob + 32 + lane], alpha * h[hb + 32 + lane]);
}

// ---- 6: per-column sum / sumsq of (agg + bias) ----
__global__ __launch_bounds__(256)
void k_colstats(const float* __restrict__ out, const float* __restrict__ bias,
                float* __restrict__ colsum, float* __restrict__ colsq, int n) {
  const int col = threadIdx.x & (OUT_F - 1);
  const int grp = threadIdx.x >> 6;                 // 0..3
  float b = bias[col];
  float s = 0.0f, q = 0.0f;
  for (int r = blockIdx.x * 4 + grp; r < n; r += gridDim.x * 4) {
    float v = out[(long)r * OUT_F + col] + b;
    s += v; q += v * v;
  }
  __shared__ float sh_s[256], sh_q[256];
  sh_s[threadIdx.x] = s; sh_q[threadIdx.x] = q;
  __syncthreads();
  if (threadIdx.x < OUT_F) {
    float ts = sh_s[threadIdx.x] + sh_s[threadIdx.x + 64] +
               sh_s[threadIdx.x + 128] + sh_s[threadIdx.x + 192];
    float tq = sh_q[threadIdx.x] + sh_q[threadIdx.x + 64] +
               sh_q[threadIdx.x + 128] + sh_q[threadIdx.x + 192];
    atomicAdd(&colsum[threadIdx.x], ts);
    atomicAdd(&colsq[threadIdx.x], tq);
  }
}

// ---- 7: BatchNorm (training stats, biased var) + ReLU, in place ----
__global__ __launch_bounds__(256)
void k_finalize(float* __restrict__ out, const float* __restrict__ bias,
                const float* __restrict__ gamma, const float* __restrict__ beta,
                const float* __restrict__ colsum, const float* __restrict__ colsq, int n) {
  long i = (long)blockIdx.x * blockDim.x + threadIdx.x;
  if (i >= (long)n * OUT_F) return;
  int col = (int)(i & (OUT_F - 1));
  float invN = 1.0f / (float)n;
  float mean = colsum[col] * invN;
  float var  = colsq[col] * invN - mean * mean;
  float v = out[i] + bias[col];
  float y = gamma[col] * (v - mean) * rsqrtf(var + BN_EPS) + beta[col];
  out[i] = fmaxf(y, 0.0f);
}

extern "C" void kernel_launch(void* const* d_in, const int* in_sizes, int n_in,
                              void* d_out, int out_size, void* d_ws, size_t ws_size,
                              hipStream_t stream) {
  const float* x        = (const float*)d_in[0];
  const int*   ei       = (const int*)  d_in[1];
  const float* emb      = (const float*)d_in[2];
  const float* W        = (const float*)d_in[3];
  const float* att_i    = (const float*)d_in[4];
  const float* att_j    = (const float*)d_in[5];
  const float* att_em_i = (const float*)d_in[6];
  const float* att_em_j = (const float*)d_in[7];
  const float* bias     = (const float*)d_in[8];
  const float* gamma    = (const float*)d_in[9];
  const float* beta     = (const float*)d_in[10];

  const int  n = in_sizes[0] / IN_F;       // 50000
  const long E = in_sizes[1] / 2;          // 1.6M
  const int* src = ei;                     // edge_index[0]
  const int* dst = ei + E;                 // edge_index[1]
  float* out = (float*)d_out;

  // workspace carve-up (floats)
  float*    h      = (float*)d_ws;                       // n*64
  float*    a_i    = h + (size_t)n * OUT_F;              // n
  float*    a_j    = a_i + n;                            // n
  float*    ev     = a_j + n;                            // E (logits, then exp)
  float*    ssum   = ev + E;                             // n
  unsigned* mxkey  = (unsigned*)(ssum + n);              // n
  float*    colsum = (float*)(mxkey + n);                // 64
  float*    colsq  = colsum + OUT_F;                     // 64

  const long totalOut  = (long)n * OUT_F;
  const int  elemBlks  = (int)((totalOut + 255) / 256);

  k_init<<<elemBlks, 256, 0, stream>>>(out, mxkey, ssum, colsum, colsq, n);

  const int gemmWaves = (n + 15) / 16;
  k_gemm_wmma<<<(gemmWaves + 7) / 8, 256, 0, stream>>>(x, W, h, n);

  k_scores<<<(n + 7) / 8, 256, 0, stream>>>(h, emb, att_i, att_j,
                                            att_em_i, att_em_j, a_i, a_j, n);

  const int eBlks = (int)((E + 255) / 256);
  k_edge_logit_max<<<eBlks, 256, 0, stream>>>(src, dst, a_i, a_j, ev, mxkey, E);
  k_edge_expsum<<<eBlks, 256, 0, stream>>>(dst, ev, mxkey, ssum, E);

  const int aggBlks = (int)((E + 7) / 8);                // wave per edge
  k_edge_agg<<<aggBlks, 256, 0, stream>>>(src, dst, ev, ssum, h, out, E);

  k_colstats<<<512, 256, 0, stream>>>(out, bias, colsum, colsq, n);
  k_finalize<<<elemBlks, 256, 0, stream>>>(out, bias, gamma, beta, colsum, colsq, n);
}